// MixGATLayer_14697378087237
// MI455X (gfx1250) — compile-verified
//
#include <hip/hip_runtime.h>
#include <hip/hip_bf16.h>

typedef float v2f __attribute__((ext_vector_type(2)));
typedef float v8f __attribute__((ext_vector_type(8)));

#define HEADS     4
#define OUT_DIM   64
#define HF        256   // HEADS*OUT_DIM
#define IN_DIM    128
#define NEG_SLOPE 0.2f
#define EPS_F     1e-16f

__device__ __forceinline__ void atomicMaxF(float* addr, float v) {
    // signed-int max for v>=0, unsigned-int min for v<0; consistent with -inf init
    if (v >= 0.0f) atomicMax((int*)addr, __float_as_int(v));
    else           atomicMin((unsigned int*)addr, __float_as_uint(v));
}

__device__ __forceinline__ void atomicAddF(float* addr, float v) {
    __hip_atomic_fetch_add(addr, v, __ATOMIC_RELAXED, __HIP_MEMORY_SCOPE_AGENT);
}

// ---- init: zero accumulator (d_out), m = -inf, s = 0 -----------------------
__global__ void k_init(float* __restrict__ out, int n_out,
                       float* __restrict__ m, float* __restrict__ s, int n_ns) {
    int i = blockIdx.x * blockDim.x + threadIdx.x;
    int stride = gridDim.x * blockDim.x;
    for (int j = i; j < n_out; j += stride) out[j] = 0.0f;
    for (int j = i; j < n_ns;  j += stride) { m[j] = -INFINITY; s[j] = 0.0f; }
}

// ---- GEMM h = x @ W via V_WMMA_F32_16X16X4_F32 -----------------------------
// one wave per 16x16 output tile; 4 waves/block cover 4 N-tiles
__global__ __launch_bounds__(128) void k_gemm(const float* __restrict__ x,
                                              const float* __restrict__ W,
                                              float* __restrict__ h, int N) {
    int wave = threadIdx.x >> 5;
    int lane = threadIdx.x & 31;
    int half = lane >> 4;     // 0: K pair {0,1}, 1: K pair {2,3}
    int l    = lane & 15;
    int m0   = blockIdx.x * 16;
    int n0   = (blockIdx.y * 4 + wave) * 16;

    int rowA = m0 + l;
    if (rowA >= N) rowA = N - 1;                 // never hit: 50000 % 16 == 0
    const float* xr = x + (size_t)rowA * IN_DIM + half * 2;
    const float* wc = W + n0 + l;

    v8f acc = {};
#pragma unroll
    for (int k = 0; k < IN_DIM; k += 4) {
        v2f a; a.x = xr[k];                    a.y = xr[k + 1];
        v2f b; b.x = wc[(k + half * 2) * HF];  b.y = wc[(k + half * 2 + 1) * HF];
        acc = __builtin_amdgcn_wmma_f32_16x16x4_f32(
                  false, a, false, b, (short)0, acc, false, false);
    }
    int col = n0 + l;
#pragma unroll
    for (int r = 0; r < 8; ++r) {
        int row = m0 + half * 8 + r;           // D layout: VGPR r -> M = r (+8)
        if (row < N) h[(size_t)row * HF + col] = acc[r];
    }
}

// ---- per-node attention logits: a_src[n,h], a_dst[n,h] ---------------------
__global__ __launch_bounds__(128) void k_attn(const float* __restrict__ h,
                                              const float* __restrict__ att_src,
                                              const float* __restrict__ att_dst,
                                              float* __restrict__ asrc,
                                              float* __restrict__ adst, int N) {
    int wave = threadIdx.x >> 5, lane = threadIdx.x & 31;
    int n = blockIdx.x * 4 + wave;
    if (n >= N) return;
    const float* hn = h + (size_t)n * HF;
#pragma unroll
    for (int hd = 0; hd < HEADS; ++hd) {
        float v0 = hn[hd * 64 + lane], v1 = hn[hd * 64 + 32 + lane];
        float ps = v0 * att_src[hd * 64 + lane] + v1 * att_src[hd * 64 + 32 + lane];
        float pd = v0 * att_dst[hd * 64 + lane] + v1 * att_dst[hd * 64 + 32 + lane];
#pragma unroll
        for (int off = 16; off > 0; off >>= 1) {
            ps += __shfl_down(ps, off, 32);
            pd += __shfl_down(pd, off, 32);
        }
        if (lane == 0) { asrc[n * HEADS + hd] = ps; adst[n * HEADS + hd] = pd; }
    }
}

// ---- edge logits + segment max over dst ------------------------------------
__global__ void k_edge_logit(const int* __restrict__ esrc, const int* __restrict__ edst,
                             const float* __restrict__ asrc, const float* __restrict__ adst,
                             float* __restrict__ e, float* __restrict__ m,
                             int E, int ET) {
    int i = blockIdx.x * blockDim.x + threadIdx.x;
    if (i >= ET) return;
    int s = (i < E) ? esrc[i] : (i - E);       // tail = self loops
    int d = (i < E) ? edst[i] : (i - E);
#pragma unroll
    for (int hd = 0; hd < HEADS; ++hd) {
        float v = asrc[s * HEADS + hd] + adst[d * HEADS + hd];
        v = (v > 0.0f) ? v : NEG_SLOPE * v;    // leaky relu
        e[(size_t)i * HEADS + hd] = v;
        atomicMaxF(&m[d * HEADS + hd], v);
    }
}

// ---- exp(e - max) + segment sum --------------------------------------------
__global__ void k_edge_exp(const int* __restrict__ edst,
                           const float* __restrict__ m,
                           float* __restrict__ e, float* __restrict__ ssum,
                           int E, int ET) {
    int i = blockIdx.x * blockDim.x + threadIdx.x;
    if (i >= ET) return;
    int d = (i < E) ? edst[i] : (i - E);
#pragma unroll
    for (int hd = 0; hd < HEADS; ++hd) {
        float v = __expf(e[(size_t)i * HEADS + hd] - m[d * HEADS + hd]);
        e[(size_t)i * HEADS + hd] = v;
        atomicAddF(&ssum[d * HEADS + hd], v);
    }
}

// ---- weighted aggregation: out[dst] += alpha * h[src], one wave per edge ---
__global__ __launch_bounds__(256) void k_agg(const int* __restrict__ esrc,
                                             const int* __restrict__ edst,
                                             const float* __restrict__ ex,
                                             const float* __restrict__ ssum,
                                             const float* __restrict__ h,
                                             float* __restrict__ out, int E, int ET) {
    int wave = threadIdx.x >> 5, lane = threadIdx.x & 31;
    int i = blockIdx.x * 8 + wave;
    if (i >= ET) return;
    int s = (i < E) ? esrc[i] : (i - E);
    int d = (i < E) ? edst[i] : (i - E);
    float alpha[HEADS];
#pragma unroll
    for (int hd = 0; hd < HEADS; ++hd)
        alpha[hd] = ex[(size_t)i * HEADS + hd] / (ssum[d * HEADS + hd] + EPS_F);
    const float* hs = h + (size_t)s * HF;
    float* od = out + (size_t)d * HF;
#pragma unroll
    for (int j = 0; j < 8; ++j) {
        int f = lane + 32 * j;                 // coalesced across the wave
        atomicAddF(&od[f], alpha[f >> 6] * hs[f]);
    }
}

// ---- epilogue: z + bias, then 0.5*z + 0.5*elu(z) ---------------------------
__global__ void k_final(float* __restrict__ out, const float* __restrict__ bias,
                        unsigned int n) {
    unsigned int i = blockIdx.x * blockDim.x + threadIdx.x;
    if (i >= n) return;
    float z = out[i] + bias[i & (HF - 1)];
    float el = (z > 0.0f) ? z : (__expf(z) - 1.0f);
    out[i] = 0.5f * z + 0.5f * el;             // BETA=0.5, C=1.0
}

extern "C" void kernel_launch(void* const* d_in, const int* in_sizes, int n_in,
                              void* d_out, int out_size, void* d_ws, size_t ws_size,
                              hipStream_t stream) {
    const float* x       = (const float*)d_in[0];
    const int*   ei      = (const int*)  d_in[1];
    const float* W       = (const float*)d_in[2];
    const float* att_src = (const float*)d_in[3];
    const float* att_dst = (const float*)d_in[4];
    const float* bias    = (const float*)d_in[5];

    int N  = in_sizes[0] / IN_DIM;   // 50000
    int E  = in_sizes[1] / 2;        // 800000
    int ET = E + N;                  // + self loops
    const int* esrc = ei;
    const int* edst = ei + E;
    float* out = (float*)d_out;

    // workspace layout (fp32): h | asrc | adst | m | ssum | e   (~68 MB)
    float* h    = (float*)d_ws;
    float* asrc = h    + (size_t)N * HF;
    float* adst = asrc + (size_t)N * HEADS;
    float* m    = adst + (size_t)N * HEADS;
    float* ssum = m    + (size_t)N * HEADS;
    float* e    = ssum + (size_t)N * HEADS;

    int n_out = N * HF;
    k_init<<<2048, 256, 0, stream>>>(out, n_out, m, ssum, N * HEADS);

    dim3 g_gemm((N + 15) / 16, 4);
    k_gemm<<<g_gemm, 128, 0, stream>>>(x, W, h, N);

    k_attn<<<(N + 3) / 4, 128, 0, stream>>>(h, att_src, att_dst, asrc, adst, N);

    k_edge_logit<<<(ET + 255) / 256, 256, 0, stream>>>(esrc, edst, asrc, adst, e, m, E, ET);
    k_edge_exp  <<<(ET + 255) / 256, 256, 0, stream>>>(edst, m, e, ssum, E, ET);
    k_agg       <<<(ET + 7) / 8,     256, 0, stream>>>(esrc, edst, e, ssum, h, out, E, ET);

    k_final<<<((unsigned)n_out + 255) / 256, 256, 0, stream>>>(out, bias, (unsigned)n_out);
}